// ConvE_32160715113083
// MI455X (gfx1250) — compile-verified
//
#include <hip/hip_runtime.h>

typedef __attribute__((ext_vector_type(2))) float v2f;
typedef __attribute__((ext_vector_type(8))) float v8f;

#define N_NODES 50000
#define N_EDGES 400000
#define DIM     200
#define NBASES  100
#define MTILES  3125   // 50000 / 16
#define NTILES  13     // ceil(200 / 16)

// ---------------------------------------------------------------------------
// Kernel 1: out = bias + emb[node_ids] @ loop_weight  (dense f32 GEMM, WMMA)
// One wave32 computes one 16x16 tile of the output, K-loop in steps of 4 via
// v_wmma_f32_16x16x4_f32.
// A 16x4 f32 layout : row = lane&15, VGPR j holds K = 2*(lane>>4) + j
// B 4x16 f32 layout : col = lane&15, VGPR j holds K = 2*(lane>>4) + j
// C/D 16x16 layout  : col = lane&15, VGPR i holds M = i + 8*(lane>>4)
// ---------------------------------------------------------------------------
__global__ __launch_bounds__(256) void rgcn_selfloop_gemm(
    const int*   __restrict__ node_ids,
    const float* __restrict__ emb,
    const float* __restrict__ W,      // loop_weight [DIM x DIM], row-major
    const float* __restrict__ bias,   // [DIM]
    float*       __restrict__ out)    // [N_NODES x DIM]
{
    const int wave = blockIdx.x * (blockDim.x >> 5) + (threadIdx.x >> 5);
    if (wave >= MTILES * NTILES) return;

    const int tn   = wave % NTILES;        // consecutive waves share the A tile
    const int tm   = wave / NTILES;
    const int lane = threadIdx.x & 31;
    const int row  = lane & 15;
    const int half = lane >> 4;

    const int m_global = tm * 16 + row;            // always < 50000
    const int a_row    = node_ids[m_global];       // h = emb[node_ids]
    const float* __restrict__ arow = emb + (size_t)a_row * DIM;

    const int  n_global = tn * 16 + row;           // output column of this lane
    const bool ncol_ok  = (n_global < DIM);
    // Clamp the B column for the padded tail tile: those lanes' C values are
    // never stored, so any finite in-bounds data is acceptable there.
    const int  bcol     = ncol_ok ? n_global : 0;

    v8f c = {};
    for (int k = 0; k < DIM; k += 4) {
        const int ka = k + 2 * half;
        v2f a, b;
        a.x = arow[ka];
        a.y = arow[ka + 1];
        b.x = W[(size_t)(ka + 0) * DIM + bcol];
        b.y = W[(size_t)(ka + 1) * DIM + bcol];
        // 8 args: (neg_a, A, neg_b, B, c_mod, C, reuse_a, reuse_b)
        c = __builtin_amdgcn_wmma_f32_16x16x4_f32(
                false, a, false, b, (short)0, c, false, false);
    }

    if (ncol_ok) {
        const float bb = bias[n_global];
        #pragma unroll
        for (int i = 0; i < 8; ++i) {
            const int m_out = tm * 16 + i + 8 * half;
            out[(size_t)m_out * DIM + n_global] = c[i] + bb;
        }
    }
}

// ---------------------------------------------------------------------------
// Kernel 2: per-(edge, block) 2x2 block-diagonal transform + norm scale +
// atomic scatter-add into out[dst].
// weight layout: [NR, NB, 2, 2]  -> block b of edge e is a contiguous float4:
//   w = { w[0][0], w[0][1], w[1][0], w[1][1] }
//   msg_o = (m0 * w[0][o] + m1 * w[1][o]) * norm
// ---------------------------------------------------------------------------
__global__ __launch_bounds__(256) void rgcn_edge_scatter(
    const int*   __restrict__ node_ids,
    const int*   __restrict__ src,
    const int*   __restrict__ dst,
    const int*   __restrict__ etype,
    const float* __restrict__ norm,
    const float* __restrict__ emb,
    const float* __restrict__ weight,  // [NR * NB * 4]
    float*       __restrict__ out)
{
    const int tid = blockIdx.x * blockDim.x + threadIdx.x;
    if (tid >= N_EDGES * NBASES) return;

    const int e = tid / NBASES;          // 100 consecutive threads -> same edge
    const int b = tid - e * NBASES;

    const int   s    = node_ids[src[e]];
    const int   t    = etype[e];
    const int   d    = dst[e];
    const float nm   = norm[e];

    const float2 m = *reinterpret_cast<const float2*>(emb + (size_t)s * DIM + 2 * b);
    const float4 w = *reinterpret_cast<const float4*>(weight + ((size_t)t * NBASES + b) * 4);

    const float o0 = (m.x * w.x + m.y * w.z) * nm;
    const float o1 = (m.x * w.y + m.y * w.w) * nm;

    float* o = out + (size_t)d * DIM + 2 * b;
    unsafeAtomicAdd(o + 0, o0);   // -> global_atomic_add_f32 (no CAS loop)
    unsafeAtomicAdd(o + 1, o1);
}

// ---------------------------------------------------------------------------
extern "C" void kernel_launch(void* const* d_in, const int* in_sizes, int n_in,
                              void* d_out, int out_size, void* d_ws, size_t ws_size,
                              hipStream_t stream) {
    const int*   node_ids    = (const int*)  d_in[0];
    const int*   src         = (const int*)  d_in[1];
    const int*   dst         = (const int*)  d_in[2];
    const int*   etype       = (const int*)  d_in[3];
    const float* norm        = (const float*)d_in[4];
    const float* emb         = (const float*)d_in[5];
    const float* weight      = (const float*)d_in[6];
    const float* loop_weight = (const float*)d_in[7];
    const float* bias        = (const float*)d_in[8];
    float*       out         = (float*)d_out;

    // Pass 1: out = bias + h @ loop_weight   (fully overwrites d_out)
    const int waves       = MTILES * NTILES;           // 40625
    const int gemm_blocks = (waves + 7) / 8;           // 8 waves per 256-thr block
    rgcn_selfloop_gemm<<<gemm_blocks, 256, 0, stream>>>(
        node_ids, emb, loop_weight, bias, out);

    // Pass 2: atomic scatter of edge messages (stream-ordered after pass 1)
    const int total       = N_EDGES * NBASES;          // 40,000,000
    const int edge_blocks = (total + 255) / 256;
    rgcn_edge_scatter<<<edge_blocks, 256, 0, stream>>>(
        node_ids, src, dst, etype, norm, emb, weight, out);
}